// MGAT_34694745817793
// MI455X (gfx1250) — compile-verified
//
#include <hip/hip_runtime.h>
#include <math.h>

// ---------------------------------------------------------------------------
// MGAT forward for MI455X (gfx1250, wave32).
// Dense GEMMs use V_WMMA_F32_16X16X4_F32 (full fp32 — GEMMs are ~11 GFLOP,
// negligible vs. the L2-resident edge gather/scatter pipeline, so we keep
// reference-grade fp32 numerics through the WMMA pipe).
// This revision removes the divergent exec-masked B loads in the GEMM K-loop:
// out-of-range columns are handled by clamping the load index (always legal)
// and multiplying by a per-lane 0/1 mask, so EXEC stays all-ones (required
// for WMMA) and the inner loop is branch-free.
// ---------------------------------------------------------------------------

#define SLOPE 0.01f

typedef __attribute__((ext_vector_type(2))) float v2f;
typedef __attribute__((ext_vector_type(8))) float v8f;

__device__ __forceinline__ float lrelu_f(float v) { return v > 0.f ? v : SLOPE * v; }

// float atomic max via int/uint punning (works for mixed signs with -inf init)
__device__ __forceinline__ void atomicMaxF(float* addr, float val) {
  if (val >= 0.f) atomicMax((int*)addr, __float_as_int(val));
  else            atomicMin((unsigned int*)addr, __float_as_uint(val));
}

// ---------------------------- fill ----------------------------------------
__global__ void k_fill(float* __restrict__ p, long n, float v) {
  long i = (long)blockIdx.x * blockDim.x + threadIdx.x;
  long st = (long)gridDim.x * blockDim.x;
  for (; i < n; i += st) p[i] = v;
}

// ---------------------------- out-degree -----------------------------------
__global__ void k_degree(const int* __restrict__ src, float* __restrict__ deg, int E) {
  int e = blockIdx.x * blockDim.x + threadIdx.x;
  if (e < E) atomicAdd(&deg[src[e]], 1.0f);
}

// ---------------------------- WMMA GEMM ------------------------------------
// C[row, ccol0+col] = epi( A[row, acol0 : acol0+K1] @ W[K1,K2] )
// 4 waves / block; each wave owns one 16x16 tile; K stepped by 4 (f32 WMMA).
// Requires: nrows % 16 == 0, K1 % 4 == 0 (true for all shapes here).
// A-layout (ISA 7.12.2, 32-bit A 16x4): lane<16 -> K{k0,k0+1}, lane>=16 -> K{k0+2,k0+3}
// C-layout: VGPR i -> row = base + i + (lane>=16 ? 8 : 0), col = lane%16
__global__ __launch_bounds__(128)
void k_gemm(const float* __restrict__ A, int lda, int acol0,
            const float* __restrict__ W,
            const float* __restrict__ bias,
            const float* __restrict__ addend, int ld_add, int add_pre,
            int act, // 0=none, 1=tanh, 2=leaky-relu
            float* __restrict__ C, int ldc, int ccol0,
            int nrows, int K1, int K2)
{
  const int wave = threadIdx.x >> 5;
  const int lane = threadIdx.x & 31;
  const int rowbase = blockIdx.x * 64 + wave * 16;
  const int colbase = blockIdx.y * 16;
  if (rowbase >= nrows) return;                 // whole-wave uniform exit

  const int mrow = rowbase + (lane & 15);
  const int hi   = lane >> 4;
  const int bcol = colbase + (lane & 15);
  const bool colok = (bcol < K2);
  const float bmask = colok ? 1.f : 0.f;        // zero contribution of OOB cols
  const int bcol_c  = colok ? bcol : (K2 - 1);  // clamped, always-legal index

  // per-lane base pointers; K-loop indexes with constant strides only
  const float* ap = A + (size_t)mrow * lda + acol0 + hi * 2;  // reads ap[0], ap[1]
  const float* bp = W + bcol_c + (size_t)(hi * 2) * K2;       // reads bp[0], bp[K2]

  v8f acc = {};
  for (int k0 = 0; k0 < K1; k0 += 4) {
    v2f a, b;
    a.x = ap[0];
    a.y = ap[1];
    b.x = bmask * bp[0];
    b.y = bmask * bp[(size_t)K2];
    acc = __builtin_amdgcn_wmma_f32_16x16x4_f32(false, a, false, b,
                                                (short)0, acc, false, false);
    ap += 4;
    bp += (size_t)4 * K2;
  }

  if (!colok) return;
  const float bv = bias ? bias[bcol] : 0.f;
#pragma unroll
  for (int i = 0; i < 8; ++i) {
    const int row = rowbase + i + hi * 8;
    float v = acc[i] + bv;
    if (addend && add_pre)  v += addend[(size_t)row * ld_add + bcol];
    if (act == 1)      v = tanhf(v);
    else if (act == 2) v = lrelu_f(v);
    if (addend && !add_pre) v += addend[(size_t)row * ld_add + bcol];
    C[(size_t)row * ldc + ccol0 + bcol] = v;
  }
}

// ------------------- edge: gated logits + segment max -----------------------
__global__ __launch_bounds__(256)
void k_edge_logits(const float* __restrict__ h, int ld, int dim,
                   const int* __restrict__ src, const int* __restrict__ dst,
                   const float* __restrict__ deg,
                   float* __restrict__ logits, float* __restrict__ mx, int E)
{
  const int wid  = (blockIdx.x * blockDim.x + threadIdx.x) >> 5;
  const int lane = threadIdx.x & 31;
  if (wid >= E) return;
  const int s = src[wid], d = dst[wid];
  const float* hs = h + (size_t)s * ld;
  const float* hd = h + (size_t)d * ld;
  float acc = 0.f;
  for (int k = lane; k < dim; k += 32)
    acc += hd[k] * lrelu_f(hs[k]);
#pragma unroll
  for (int off = 16; off > 0; off >>= 1)
    acc += __shfl_xor(acc, off, 32);
  if (lane == 0) {
    const float inner = acc;
    const float t    = inner * rsqrtf(deg[s]);     // deg>=1 for any used src
    const float gate = 1.f / (1.f + expf(-t));
    const float lg   = inner * gate;
    logits[wid] = lg;
    atomicMaxF(&mx[d], lg);
  }
}

// ------------------- edge: exp + segment denom ------------------------------
__global__ void k_edge_exp(const float* __restrict__ logits,
                           const int* __restrict__ dst,
                           const float* __restrict__ mx,
                           float* __restrict__ eexp, float* __restrict__ denom, int E)
{
  int e = blockIdx.x * blockDim.x + threadIdx.x;
  if (e >= E) return;
  const int d = dst[e];
  const float ex = expf(logits[e] - mx[d]);
  eexp[e] = ex;
  atomicAdd(&denom[d], ex);
}

// ------------------- edge: attention-weighted scatter-add -------------------
__global__ __launch_bounds__(256)
void k_edge_scatter(const float* __restrict__ h, int ld, int dim,
                    const int* __restrict__ src, const int* __restrict__ dst,
                    const float* __restrict__ eexp, const float* __restrict__ denom,
                    float* __restrict__ gout, int E)
{
  const int wid  = (blockIdx.x * blockDim.x + threadIdx.x) >> 5;
  const int lane = threadIdx.x & 31;
  if (wid >= E) return;
  const int s = src[wid], d = dst[wid];
  const float attn = eexp[wid] / (denom[d] + 1e-16f);
  const float* hs = h + (size_t)s * ld;
  float* od = gout + (size_t)d * dim;
  for (int k = lane; k < dim; k += 32)
    atomicAdd(&od[k], hs[k] * attn);
}

// ------------------- row: (+bias) -> l2norm -> (lrelu) ----------------------
__global__ __launch_bounds__(256)
void k_row_finish(const float* __restrict__ in, int ldi,
                  const float* __restrict__ bias,
                  float* __restrict__ out, int ldo,
                  int dim, int nrows, int do_lrelu)
{
  const int row  = (blockIdx.x * blockDim.x + threadIdx.x) >> 5;
  const int lane = threadIdx.x & 31;
  if (row >= nrows) return;
  const float* ip = in + (size_t)row * ldi;
  float vals[8];                                  // dim <= 256 -> <= 8/lane
  float ss = 0.f;
  int cnt = 0;
  for (int k = lane; k < dim; k += 32) {
    float v = ip[k] + (bias ? bias[k] : 0.f);
    vals[cnt++] = v;
    ss += v * v;
  }
#pragma unroll
  for (int off = 16; off > 0; off >>= 1)
    ss += __shfl_xor(ss, off, 32);
  const float scale = 1.f / fmaxf(sqrtf(ss), 1e-12f);
  float* op = out + (size_t)row * ldo;
  cnt = 0;
  for (int k = lane; k < dim; k += 32) {
    float v = vals[cnt++] * scale;
    if (do_lrelu) v = lrelu_f(v);
    op[k] = v;
  }
}

// ------------------- final average ------------------------------------------
__global__ void k_avg(const float* __restrict__ a, const float* __restrict__ b,
                      float* __restrict__ o, long n)
{
  long i = (long)blockIdx.x * blockDim.x + threadIdx.x;
  long st = (long)gridDim.x * blockDim.x;
  for (; i < n; i += st) o[i] = 0.5f * (a[i] + b[i]);
}

// ===========================================================================
// Host-side orchestration
// ===========================================================================
static inline void gemm(hipStream_t st, const float* A, int lda, int acol0,
                        const float* W, const float* bias,
                        const float* addend, int ld_add, int add_pre, int act,
                        float* C, int ldc, int ccol0, int nrows, int K1, int K2)
{
  dim3 grid((nrows + 63) / 64, (K2 + 15) / 16);
  k_gemm<<<grid, 128, 0, st>>>(A, lda, acol0, W, bias, addend, ld_add, add_pre,
                               act, C, ldc, ccol0, nrows, K1, K2);
}

static void run_gat(hipStream_t st, const float* X, int ldx, int K1,
                    const float* W, const float* bias, int dim,
                    const int* src, const int* dst, int E, int N,
                    float* hbuf, float* gout, float* logits, float* eexp,
                    float* mx, float* denom, const float* deg,
                    float* dest, int lddest)
{
  // h = X @ W
  gemm(st, X, ldx, 0, W, nullptr, nullptr, 0, 0, 0, hbuf, dim, 0, N, K1, dim);
  k_fill<<<2048, 256, 0, st>>>(gout, (long)N * dim, 0.f);
  k_fill<<<256, 256, 0, st>>>(mx, N, -INFINITY);
  k_fill<<<256, 256, 0, st>>>(denom, N, 0.f);
  const int eblk = (E + 7) / 8;                    // wave-per-edge, 8 waves/block
  k_edge_logits<<<eblk, 256, 0, st>>>(hbuf, dim, dim, src, dst, deg, logits, mx, E);
  k_edge_exp<<<(E + 255) / 256, 256, 0, st>>>(logits, dst, mx, eexp, denom, E);
  k_edge_scatter<<<eblk, 256, 0, st>>>(hbuf, dim, dim, src, dst, eexp, denom, gout, E);
  // dest = lrelu(l2norm(gout + bias))
  k_row_finish<<<(N + 7) / 8, 256, 0, st>>>(gout, dim, bias, dest, lddest, dim, N, 1);
}

static void run_branch(hipStream_t st, const float* feat, int ldf, int fcol0, int fdim,
                       const float* const* p, int lat, const float* id_emb,
                       const int* src, const int* dst, int E, int N,
                       float* xbuf, float* hbuf, float* gout, float* xhat,
                       float* logits, float* eexp, float* mx, float* denom,
                       const float* deg, float* rep /* [N,128] slice of d_out */)
{
  const float *mlp_w = p[0],  *mlp_b = p[1],  *c1_w = p[2],  *c1_b = p[3],
              *l1_w  = p[4],  *l1_b  = p[5],  *g1_w = p[6],  *g1_b = p[7],
              *c2_w  = p[8],  *c2_b  = p[9],  *l2_w = p[10], *l2_b = p[11],
              *g2_w  = p[12], *g2_b  = p[13];

  // x = l2norm(tanh(feat_part @ mlp_w + mlp_b))
  gemm(st, feat, ldf, fcol0, mlp_w, mlp_b, nullptr, 0, 0, /*tanh*/1,
       xbuf, lat, 0, N, fdim, lat);
  k_row_finish<<<(N + 7) / 8, 256, 0, st>>>(xbuf, lat, nullptr, xbuf, lat, lat, N, 0);

  // h = lrelu(GraphGAT(x, c1))  -> hbuf [N,lat]
  run_gat(st, xbuf, lat, lat, c1_w, c1_b, lat, src, dst, E, N,
          hbuf, gout, logits, eexp, mx, denom, deg, hbuf, lat);

  // x_hat = lrelu(x @ l1_w + l1_b) + id_emb
  gemm(st, xbuf, lat, 0, l1_w, l1_b, id_emb, 64, /*post*/0, 2, xhat, 64, 0, N, lat, 64);
  // x1 = lrelu(h @ g1_w + g1_b + x_hat)  -> rep[:, 0:64]
  gemm(st, hbuf, lat, 0, g1_w, g1_b, xhat, 64, /*pre*/1, 2, rep, 128, 0, N, lat, 64);

  // h2 = lrelu(GraphGAT(x1, c2)) -> hbuf [N,64]
  run_gat(st, rep, 128, 64, c2_w, c2_b, 64, src, dst, E, N,
          hbuf, gout, logits, eexp, mx, denom, deg, hbuf, 64);

  // x_hat2 = lrelu(x1 @ l2_w + l2_b) + id_emb
  gemm(st, rep, 128, 0, l2_w, l2_b, id_emb, 64, 0, 2, xhat, 64, 0, N, 64, 64);
  // x2 = lrelu(h2 @ g2_w + g2_b + x_hat2) -> rep[:, 64:128]
  gemm(st, hbuf, 64, 0, g2_w, g2_b, xhat, 64, 1, 2, rep, 128, 64, N, 64, 64);
}

extern "C" void kernel_launch(void* const* d_in, const int* in_sizes, int n_in,
                              void* d_out, int out_size, void* d_ws, size_t ws_size,
                              hipStream_t stream)
{
  const float* feat   = (const float*)d_in[0];
  const int*   eidx   = (const int*)d_in[1];
  const float* id_emb = (const float*)d_in[2];

  const int N = in_sizes[2] / 64;        // id_embedding [N,64]
  const int E = in_sizes[1] / 2;         // edge_index [2,E]
  const int FTOT = in_sizes[0] / N;      // 192 = V_FEAT + T_FEAT
  const int* src = eidx;
  const int* dst = eidx + E;

  const float* vprm[14];
  const float* tprm[14];
  for (int i = 0; i < 14; ++i) {
    vprm[i] = (const float*)d_in[3 + i];
    tprm[i] = (const float*)d_in[17 + i];
  }

  // ---- workspace carve (floats): xbuf/hbuf/gout N*256, xhat N*64,
  //      logits/eexp E, mx/denom/deg N  -> ~104 MB total
  float* xbuf   = (float*)d_ws;
  float* hbuf   = xbuf + (size_t)N * 256;
  float* gout   = hbuf + (size_t)N * 256;
  float* xhat   = gout + (size_t)N * 256;
  float* logits = xhat + (size_t)N * 64;
  float* eexp   = logits + (size_t)E;
  float* mx     = eexp + (size_t)E;
  float* denom  = mx + (size_t)N;
  float* deg    = denom + (size_t)N;

  float* out  = (float*)d_out;
  float* vrep = out + (size_t)N * 128;
  float* trep = out + (size_t)N * 128 * 2;

  // out-degree (shared by all 4 GAT layers)
  k_fill<<<256, 256, 0, stream>>>(deg, N, 0.f);
  k_degree<<<(E + 255) / 256, 256, 0, stream>>>(src, deg, E);

  // V branch: feat[:, 0:128], lat 256
  run_branch(stream, feat, FTOT, 0, 128, vprm, 256, id_emb, src, dst, E, N,
             xbuf, hbuf, gout, xhat, logits, eexp, mx, denom, deg, vrep);
  // T branch: feat[:, 128:192], lat 100
  run_branch(stream, feat, FTOT, 128, 64, tprm, 100, id_emb, src, dst, E, N,
             xbuf, hbuf, gout, xhat, logits, eexp, mx, denom, deg, trep);

  // representation = (v_rep + t_rep) / 2
  k_avg<<<2048, 256, 0, stream>>>(vrep, trep, out, (long)N * 128);
}